// MemNet_8478265442480
// MI455X (gfx1250) — compile-verified
//
#include <hip/hip_runtime.h>
#include <hip/hip_bf16.h>
#include <math.h>

// Problem constants
constexpr int kB = 128;
constexpr int kS = 2048;
constexpr int kT = 4;
constexpr int kD = 300;
constexpr int kDP = 304;    // kD padded to 19 full 16-wide N tiles
constexpr int kC = 3;       // N_CLASSES
constexpr int kHops = 3;
constexpr int kMT = 32;     // (kB*kT)/16 = 512/16 M-tiles (exact)
constexpr int kNT = 19;     // 304/16 N-tiles (exact, padded)

typedef __attribute__((ext_vector_type(2))) float v2f;
typedef __attribute__((ext_vector_type(8))) float v8f;

__device__ __forceinline__ float wave_reduce_sum(float v) {
    v += __shfl_xor(v, 16, 32);
    v += __shfl_xor(v, 8, 32);
    v += __shfl_xor(v, 4, 32);
    v += __shfl_xor(v, 2, 32);
    v += __shfl_xor(v, 1, 32);
    return v;
}

// ---------------------------------------------------------------------------
// 0) Build transposed+padded weights once:
//    Wt[col][k] = W_tr[k][col] for col<300 else 0   (col in [0,304), k in [0,300))
//    btr_pad[col] = b_tr[col] for col<300 else 0
//    => B-fragment (k,col),(k+1,col) becomes one contiguous b64 load
// ---------------------------------------------------------------------------
__global__ void pad_weights_kernel(const float* __restrict__ W_tr,
                                   const float* __restrict__ b_tr,
                                   float* __restrict__ Wt,
                                   float* __restrict__ btr_pad) {
    const int i = blockIdx.x * blockDim.x + threadIdx.x;
    const int nW = kDP * kD;
    if (i < nW) {
        const int col = i / kD, k = i % kD;
        Wt[i] = (col < kD) ? W_tr[k * kD + col] : 0.f;
    } else if (i < nW + kDP) {
        const int col = i - nW;
        btr_pad[col] = (col < kD) ? b_tr[col] : 0.f;
    }
}

// ---------------------------------------------------------------------------
// 1) p[b,s] = emb[inputs[b,s],:] . Wa   (hop-invariant attention logit part)
// ---------------------------------------------------------------------------
__global__ void compute_p_kernel(const int* __restrict__ inputs,
                                 const float* __restrict__ emb,
                                 const float* __restrict__ W_att,  // Wa = [0..299]
                                 float* __restrict__ p) {
    const int wave = threadIdx.x >> 5;
    const int lane = threadIdx.x & 31;
    const int row = blockIdx.x * 8 + wave;          // row in [0, kB*kS)
    const float* er = emb + (size_t)inputs[row] * kD;
    float acc = 0.f;
    for (int k = lane; k < kD; k += 32) acc += er[k] * W_att[k];
    acc = wave_reduce_sum(acc);
    if (lane == 0) p[row] = acc;
}

// ---------------------------------------------------------------------------
// 2) te0 = emb[targets] (padded rows); u0 = mean_t te0
// ---------------------------------------------------------------------------
__global__ void init_u_kernel(const int* __restrict__ targets,
                              const float* __restrict__ emb,
                              float* __restrict__ te,   // [kB*kT][kDP]
                              float* __restrict__ u) {  // [kB][kD]
    const int b = blockIdx.x;
    for (int i = threadIdx.x; i < kT * kD; i += blockDim.x) {
        const int t = i / kD, d = i % kD;
        te[(b * kT + t) * kDP + d] = emb[(size_t)targets[b * kT + t] * kD + d];
    }
    for (int d = threadIdx.x; d < kD; d += blockDim.x) {
        float acc = 0.f;
        for (int t = 0; t < kT; ++t)
            acc += emb[(size_t)targets[b * kT + t] * kD + d];
        u[b * kD + d] = acc * (1.f / kT);
    }
}

// ---------------------------------------------------------------------------
// 3a) per batch: q = u.Wu + b_att; ew[s] = exp(tanh(p[s]+q)); asum = sum(ew)
//     tanh output in (-1,1) => exp overflow-safe, no max pass needed
// ---------------------------------------------------------------------------
__global__ void attn_softmax_kernel(const float* __restrict__ p,
                                    const float* __restrict__ u,
                                    const float* __restrict__ W_att,  // Wu at [kD..2kD)
                                    const float* __restrict__ b_att,
                                    float* __restrict__ ew,
                                    float* __restrict__ asum) {
    __shared__ float red[256];
    __shared__ float q_sh;
    const int b = blockIdx.x, t = threadIdx.x;

    float local = 0.f;
    for (int d = t; d < kD; d += 256) local += u[b * kD + d] * W_att[kD + d];
    red[t] = local;
    __syncthreads();
    for (int off = 128; off >= 1; off >>= 1) {
        if (t < off) red[t] += red[t + off];
        __syncthreads();
    }
    if (t == 0) q_sh = red[0] + b_att[0];
    __syncthreads();
    const float q = q_sh;

    float lsum = 0.f;
    for (int s = t; s < kS; s += 256) {
        const float g = tanhf(p[b * kS + s] + q);
        const float e = __expf(g);
        ew[b * kS + s] = e;
        lsum += e;
    }
    red[t] = lsum;
    __syncthreads();
    for (int off = 128; off >= 1; off >>= 1) {
        if (t < off) red[t] += red[t + off];
        __syncthreads();
    }
    if (t == 0) asum[b] = red[0];
}

// ---------------------------------------------------------------------------
// 3b) vec[b,:] = (1/asum) * sum_s ew[s] * emb[inputs[b,s],:]
//     one block/batch; thread t owns dims t and 256+min(t,43); the clamp makes
//     the second load always in-bounds so the hot loop is branch-free
//     (lanes t>=44 accumulate a duplicate that is simply never stored).
// ---------------------------------------------------------------------------
__global__ void weighted_sum_kernel(const int* __restrict__ inputs,
                                    const float* __restrict__ emb,
                                    const float* __restrict__ ew,
                                    const float* __restrict__ asum,
                                    float* __restrict__ vec) {   // [kB][kDP]
    const int b = blockIdx.x, t = threadIdx.x;   // 256 threads
    const int t2 = 256 + (t < kD - 256 ? t : kD - 257);  // in [256, 299]
    const int* idx = inputs + b * kS;
    const float* e = ew + b * kS;
    float acc0 = 0.f, acc1 = 0.f;
#pragma unroll 4
    for (int s = 0; s < kS; ++s) {
        const float* row = emb + (size_t)idx[s] * kD;
        const float w = e[s];
        acc0 += w * row[t];
        acc1 += w * row[t2];
    }
    const float inv = 1.f / asum[b];
    vec[b * kDP + t] = acc0 * inv;
    if (t < kD - 256) vec[b * kDP + t2] = acc1 * inv;
    if (t < kDP - kD) vec[b * kDP + kD + t] = 0.f;   // deterministic pad cols
}

// ---------------------------------------------------------------------------
// 3c) te_out = vec[b] + te_in @ W_tr + b_tr  via V_WMMA_F32_16X16X4_F32
//     M=512 (32 tiles), N=304 padded (19 tiles), K=300 (75 steps of 4).
//     One wave per 16x16 tile, 608 waves exactly. Fully branch-free:
//     EXEC stays all-1s (WMMA requirement), pad columns compute to 0.
// ---------------------------------------------------------------------------
__global__ void te_update_wmma_kernel(const float* __restrict__ te_in,  // [512][kDP]
                                      const float* __restrict__ Wt,     // [kDP][kD]
                                      const float* __restrict__ btr,    // [kDP]
                                      const float* __restrict__ vec,    // [kB][kDP]
                                      float* __restrict__ te_out) {     // [512][kDP]
    const int waveId = (int)((blockIdx.x * blockDim.x + threadIdx.x) >> 5);
    const int lane = threadIdx.x & 31;
    const int mt = waveId / kNT;
    const int nt = waveId % kNT;

    const int halfSel = lane >> 4;         // 0: lanes 0-15, 1: lanes 16-31
    const int l16 = lane & 15;
    const int arow = mt * 16 + l16;        // A row supplied by this lane
    const int bcol = nt * 16 + l16;        // B/D column supplied by this lane
    const float* aptr = te_in + arow * kDP;   // 8B-aligned at even kk
    const float* bptr = Wt + bcol * kD;       // contiguous (k,k+1) pairs

    v8f c = {};
    for (int k0 = 0; k0 < kD; k0 += 4) {
        const int kk = k0 + (halfSel << 1);   // this lane supplies K = kk, kk+1
        const v2f a  = *(const v2f*)(aptr + kk);
        const v2f bf = *(const v2f*)(bptr + kk);
        c = __builtin_amdgcn_wmma_f32_16x16x4_f32(
                /*neg_a=*/false, a, /*neg_b=*/false, bf,
                /*c_mod=*/(short)0, c, /*reuse_a=*/false, /*reuse_b=*/false);
    }

    const float badd = btr[bcol];
#pragma unroll
    for (int r = 0; r < 8; ++r) {
        const int mrow = mt * 16 + r + (halfSel << 3);  // C/D: M=r (+8 for hi half)
        const int bb = mrow >> 2;                       // batch = row / kT
        te_out[mrow * kDP + bcol] = c[r] + vec[bb * kDP + bcol] + badd;
    }
}

// ---------------------------------------------------------------------------
// 3d) u[b,:] = mean_t te[b,t,:]
// ---------------------------------------------------------------------------
__global__ void mean_u_kernel(const float* __restrict__ te,  // [512][kDP]
                              float* __restrict__ u) {       // [kB][kD]
    const int b = blockIdx.x;
    for (int d = threadIdx.x; d < kD; d += blockDim.x) {
        const float s = te[(b * kT + 0) * kDP + d] + te[(b * kT + 1) * kDP + d] +
                        te[(b * kT + 2) * kDP + d] + te[(b * kT + 3) * kDP + d];
        u[b * kD + d] = s * 0.25f;
    }
}

// ---------------------------------------------------------------------------
// 4) out[b,c] = u[b,:].W_out[:,c] + b_out[c]; one wave per class
// ---------------------------------------------------------------------------
__global__ void out_kernel(const float* __restrict__ u,
                           const float* __restrict__ W_out,
                           const float* __restrict__ b_out,
                           float* __restrict__ out) {
    const int b = blockIdx.x;
    const int cls = threadIdx.x >> 5;
    const int lane = threadIdx.x & 31;
    if (cls >= kC) return;
    float acc = 0.f;
    for (int d = lane; d < kD; d += 32) acc += u[b * kD + d] * W_out[d * kC + cls];
    acc = wave_reduce_sum(acc);
    if (lane == 0) out[b * kC + cls] = acc + b_out[cls];
}

// ---------------------------------------------------------------------------
extern "C" void kernel_launch(void* const* d_in, const int* in_sizes, int n_in,
                              void* d_out, int out_size, void* d_ws, size_t ws_size,
                              hipStream_t stream) {
    const int*   inputs  = (const int*)d_in[0];
    const int*   targets = (const int*)d_in[1];
    const float* emb     = (const float*)d_in[2];
    const float* W_att   = (const float*)d_in[3];
    const float* b_att   = (const float*)d_in[4];
    const float* W_tr    = (const float*)d_in[5];
    const float* b_tr    = (const float*)d_in[6];
    const float* W_out   = (const float*)d_in[7];
    const float* b_out   = (const float*)d_in[8];
    float* out = (float*)d_out;

    // workspace layout (floats): ~4.0 MB total
    float* ws      = (float*)d_ws;
    float* p       = ws;                        // kB*kS
    float* ew      = p + kB * kS;               // kB*kS
    float* asum    = ew + kB * kS;              // kB
    float* te_a    = asum + kB;                 // kB*kT*kDP
    float* te_b    = te_a + kB * kT * kDP;      // kB*kT*kDP
    float* u       = te_b + kB * kT * kDP;      // kB*kD
    float* vec     = u + kB * kD;               // kB*kDP
    float* Wt      = vec + kB * kDP;            // kDP*kD
    float* btr_pad = Wt + kDP * kD;             // kDP

    const int nPad = kDP * kD + kDP;
    pad_weights_kernel<<<(nPad + 255) / 256, 256, 0, stream>>>(W_tr, b_tr, Wt, btr_pad);
    compute_p_kernel<<<(kB * kS) / 8, 256, 0, stream>>>(inputs, emb, W_att, p);
    init_u_kernel<<<kB, 256, 0, stream>>>(targets, emb, te_a, u);

    float* te_src = te_a;
    float* te_dst = te_b;
    for (int hop = 0; hop < kHops; ++hop) {
        attn_softmax_kernel<<<kB, 256, 0, stream>>>(p, u, W_att, b_att, ew, asum);
        weighted_sum_kernel<<<kB, 256, 0, stream>>>(inputs, emb, ew, asum, vec);
        // 608 tiles exactly: 76 blocks x 8 waves
        te_update_wmma_kernel<<<(kMT * kNT) / 8, 256, 0, stream>>>(
            te_src, Wt, btr_pad, vec, te_dst);
        mean_u_kernel<<<kB, 256, 0, stream>>>(te_dst, u);
        float* tmp = te_src; te_src = te_dst; te_dst = tmp;
    }
    out_kernel<<<kB, 96, 0, stream>>>(u, W_out, b_out, out);
}